// Model_28681791602777
// MI455X (gfx1250) — compile-verified
//
#include <hip/hip_runtime.h>

// out[i] = verified_id[i*ndt + accept_lens[i] - 1], i in [0, bs)
//
// Memory-bound gather: ~160 MB total traffic (128 MB verified_id region at
// cacheline granularity + 16 MB lens + 16 MB out) -> ~7 us at 23.3 TB/s.
// Strategy: 4 rows/thread, b128 NT load of lens, 4x b32 NT gather loads,
// b128 NT store. NT hints keep the zero-reuse stream from thrashing L2.
// NDT is a compile-time constant (8) so gather addresses are shift+add in
// 32-bit saddr+voffset form instead of 64-bit v_mul_u64 chains.

typedef int   v4i __attribute__((ext_vector_type(4)));
typedef float v4f __attribute__((ext_vector_type(4)));

template <int NDT>
__global__ __launch_bounds__(256) void gather_accept_ndt(
    const float* __restrict__ verified_id,
    const int*   __restrict__ accept_lens,
    float*       __restrict__ out,
    int bs)
{
    int i = (blockIdx.x * blockDim.x + threadIdx.x) * 4;

    if (__builtin_expect(i + 3 < bs, 1)) {
        // Coalesced 16B/lane load of 4 accept lengths (non-temporal b128).
        v4i lens = __builtin_nontemporal_load(
            reinterpret_cast<const v4i*>(accept_lens + i));

        // 32-bit offsets: NDT*bs*4 bytes = 128 MB < 4 GB, so unsigned math
        // is exact and the compiler can use saddr + 32-bit voffset addressing.
        unsigned b = (unsigned)i * NDT;
        v4f r;
        r.x = __builtin_nontemporal_load(verified_id + (b + 0 * NDT + (unsigned)(lens.x - 1)));
        r.y = __builtin_nontemporal_load(verified_id + (b + 1 * NDT + (unsigned)(lens.y - 1)));
        r.z = __builtin_nontemporal_load(verified_id + (b + 2 * NDT + (unsigned)(lens.z - 1)));
        r.w = __builtin_nontemporal_load(verified_id + (b + 3 * NDT + (unsigned)(lens.w - 1)));

        // Coalesced 16B/lane non-temporal b128 store.
        __builtin_nontemporal_store(r, reinterpret_cast<v4f*>(out + i));
    } else {
        // Tail: never executed for BS = 4M (grid covers it exactly).
        #pragma clang loop unroll(disable) vectorize(disable)
        for (; i < bs; ++i) {
            int len = accept_lens[i];
            out[i] = verified_id[(size_t)i * NDT + (len - 1)];
        }
    }
}

// Generic fallback for unexpected ndt (kept scalar-simple; not the hot path).
__global__ __launch_bounds__(256) void gather_accept_generic(
    const float* __restrict__ verified_id,
    const int*   __restrict__ accept_lens,
    float*       __restrict__ out,
    int bs, int ndt)
{
    int i = blockIdx.x * blockDim.x + threadIdx.x;
    if (i < bs) {
        int len = accept_lens[i];
        out[i] = verified_id[(size_t)i * ndt + (len - 1)];
    }
}

extern "C" void kernel_launch(void* const* d_in, const int* in_sizes, int n_in,
                              void* d_out, int out_size, void* d_ws, size_t ws_size,
                              hipStream_t stream) {
    const float* verified_id = (const float*)d_in[0];
    const int*   accept_lens = (const int*)d_in[1];
    float*       out         = (float*)d_out;

    const int bs  = in_sizes[1];               // 4,194,304
    const int ndt = in_sizes[0] / in_sizes[1]; // 8

    if (ndt == 8) {
        const int threads = 256;               // 8 wave32s per block
        const int total_threads = (bs + 3) / 4;
        const int blocks = (total_threads + threads - 1) / threads;
        gather_accept_ndt<8><<<blocks, threads, 0, stream>>>(
            verified_id, accept_lens, out, bs);
    } else {
        const int threads = 256;
        const int blocks = (bs + threads - 1) / threads;
        gather_accept_generic<<<blocks, threads, 0, stream>>>(
            verified_id, accept_lens, out, bs, ndt);
    }
}